// RGT_3298534884300
// MI455X (gfx1250) — compile-verified
//
#include <hip/hip_runtime.h>

// ============================================================================
// Relational Graph Transformer for MI455X (gfx1250, wave32, WMMA).
// - All dense contractions: V_WMMA_F32_16X16X4_F32 (full f32 on matrix pipe;
//   total GEMM flops ~123 GF are negligible vs the L2-resident edge phase, so
//   no precision downconversion is warranted).
// - GEMMs templated on (K,Nc): compile-time strides kill the runtime 64-bit
//   address-math chains seen in round 1.
// - B panels staged in LDS pair-major so each lane's {k,k+1} B fragment is one
//   ds_load_b64; shared by all 8 waves of the block (8x less B traffic).
// - Each wave computes a 16x32 tile (2 accumulators reuse one A fragment).
// - Semantic attention einsum+tanh+dot+mean fully fused: WMMA tile reduced
//   in-register, one atomicAdd per wave; [S,N,T,HS] (~410 MB) never exists.
// ============================================================================

typedef float v2f __attribute__((ext_vector_type(2)));
typedef float v8f __attribute__((ext_vector_type(8)));

static constexpr int Nn  = 50000;
static constexpr int Ee  = 400000;
static constexpr int Ff  = 768;
static constexpr int Dd  = 128;
static constexpr int Tt  = 2;
static constexpr int Hh  = 2;
static constexpr int Ss  = 8;
static constexpr int HSs = 128;

#define DEV_INLINE __device__ __forceinline__

// ---------------------------------------------------------------------------
// Tiled WMMA GEMM:  C[M,Nc] = act( A[M,K] @ B[K,Nc] + bias + addmat )
// 8 waves/block; wave w owns rows [tileM,tileM+16) x cols [tileN,tileN+32).
// B staged in LDS pair-major: (k,c) -> bsh[(k>>1)*64 + c*2 + (k&1)], so the
// per-lane B fragment {B[k+2h][c], B[k+2h+1][c]} is one aligned ds_load_b64.
// f32 WMMA fragment layout (ISA 7.12.2):
//   A 16x4 : lanes 0-15 = rows; vgpr0={K0|K2}, vgpr1={K1|K3} per lane-half
//   B 4x16 : lanes = cols; vgpr0={K0|K2}, vgpr1={K1|K3} per lane-half
//   C 16x16: vgpr r = rows r (lanes 0-15) / r+8 (lanes 16-31)
// No early returns: all waves reach every barrier; WMMA runs with full EXEC
// (garbage rows clamped on load, masked on store).
// ---------------------------------------------------------------------------
template<int K, int Nc>
__global__ __launch_bounds__(256)
void gemm_wmma_f32(const float* __restrict__ A, const float* __restrict__ B,
                   const float* __restrict__ bias, const float* __restrict__ addmat,
                   float* __restrict__ C, int M, int act)
{
    constexpr int KC = (K < 256) ? K : 256;          // K rows staged per chunk
    __shared__ float bsh[(KC / 2) * 64];             // <= 32 KB

    const int tid  = threadIdx.x;
    const int lane = tid & 31, wv = tid >> 5;
    const int half = lane >> 4, l = lane & 15;
    const int tileM = (blockIdx.x * 8 + wv) * 16;
    const int tileN = blockIdx.y * 32;

    int arow = tileM + l; if (arow >= M) arow = M - 1;       // clamp loads
    const float* Ap = A + (size_t)arow * K + 2 * half;

    v8f acc0 = {0.f,0.f,0.f,0.f,0.f,0.f,0.f,0.f};
    v8f acc1 = {0.f,0.f,0.f,0.f,0.f,0.f,0.f,0.f};

    for (int kc = 0; kc < K; kc += KC) {
        __syncthreads();                              // previous chunk consumed
        for (int i = tid; i < KC * 32; i += 256) {    // coalesced B stage
            int k = i >> 5, c = i & 31;
            bsh[(k >> 1) * 64 + c * 2 + (k & 1)] = B[(size_t)(kc + k) * Nc + tileN + c];
        }
        __syncthreads();
        #pragma unroll 4
        for (int k = 0; k < KC; k += 4) {
            v2f a;
            a.x = Ap[kc + k];
            a.y = Ap[kc + k + 1];
            const float* bp = &bsh[((k >> 1) + half) * 64];
            v2f b0 = *(const v2f*)(bp + l * 2);            // ds_load_b64
            v2f b1 = *(const v2f*)(bp + 32 + l * 2);       // ds_load_b64
            acc0 = __builtin_amdgcn_wmma_f32_16x16x4_f32(false, a, false, b0,
                                                         (short)0, acc0, false, false);
            acc1 = __builtin_amdgcn_wmma_f32_16x16x4_f32(false, a, false, b1,
                                                         (short)0, acc1, false, false);
        }
    }

    const float bv0 = bias ? bias[tileN + l]      : 0.f;
    const float bv1 = bias ? bias[tileN + 16 + l] : 0.f;
    #pragma unroll
    for (int r = 0; r < 8; ++r) {
        int row = tileM + r + 8 * half;
        if (row < M) {
            size_t i0 = (size_t)row * Nc + tileN + l;
            float v0 = acc0[r] + bv0;
            float v1 = acc1[r] + bv1;
            if (addmat) { v0 += addmat[i0]; v1 += addmat[i0 + 16]; }
            if (act) {
                v0 = v0 > 0.f ? v0 : 0.01f * v0;
                v1 = v1 > 0.f ? v1 : 0.01f * v1;
            }
            C[i0] = v0;
            C[i0 + 16] = v1;
        }
    }
}

// ---------------------------------------------------------------------------
// Fused semantic attention: wsum[s,t] += sum_n sum_h tanh((z_t @ Wsem1[s])[n,h]
//                                        + bsem1[s,h]) * wsem2[s,h]
// Same dual-tile + LDS-staged-B scheme; result reduced in-register to one
// atomicAdd per wave. Col-chunked (4 x 32 cols), K = Dd = 128.
// ---------------------------------------------------------------------------
__global__ __launch_bounds__(256)
void sem_wsum_k(const float* __restrict__ z, const float* __restrict__ Wsem1,
                const float* __restrict__ bsem1, const float* __restrict__ wsem2,
                float* __restrict__ wsum)
{
    __shared__ float bsh[(Dd / 2) * 64];              // 16 KB

    const int tid  = threadIdx.x;
    const int lane = tid & 31, wv = tid >> 5;
    const int half = lane >> 4, l = lane & 15;
    const int st   = blockIdx.y;                      // s*2 + t
    const int s    = st >> 1, t = st & 1;
    const int tileM = (blockIdx.x * 8 + wv) * 16;

    int arow = tileM + l; if (arow >= Nn) arow = Nn - 1;
    const float* Ap    = z + (size_t)arow * (Tt * Dd) + t * Dd + 2 * half;
    const float* Bbase = Wsem1 + (size_t)s * Dd * HSs;

    float partial = 0.f;
    for (int ct = 0; ct < HSs / 32; ++ct) {
        __syncthreads();
        for (int i = tid; i < Dd * 32; i += 256) {
            int k = i >> 5, c = i & 31;
            bsh[(k >> 1) * 64 + c * 2 + (k & 1)] = Bbase[(size_t)k * HSs + ct * 32 + c];
        }
        __syncthreads();

        v8f acc0 = {0.f,0.f,0.f,0.f,0.f,0.f,0.f,0.f};
        v8f acc1 = {0.f,0.f,0.f,0.f,0.f,0.f,0.f,0.f};
        #pragma unroll 4
        for (int k = 0; k < Dd; k += 4) {
            v2f a;
            a.x = Ap[k];
            a.y = Ap[k + 1];
            const float* bp = &bsh[((k >> 1) + half) * 64];
            v2f b0 = *(const v2f*)(bp + l * 2);
            v2f b1 = *(const v2f*)(bp + 32 + l * 2);
            acc0 = __builtin_amdgcn_wmma_f32_16x16x4_f32(false, a, false, b0,
                                                         (short)0, acc0, false, false);
            acc1 = __builtin_amdgcn_wmma_f32_16x16x4_f32(false, a, false, b1,
                                                         (short)0, acc1, false, false);
        }

        const int col0 = ct * 32 + l, col1 = col0 + 16;
        const float bs0 = bsem1[s * HSs + col0], w20 = wsem2[s * HSs + col0];
        const float bs1 = bsem1[s * HSs + col1], w21 = wsem2[s * HSs + col1];
        #pragma unroll
        for (int r = 0; r < 8; ++r) {
            int row = tileM + r + 8 * half;
            if (row < Nn) {
                partial += tanhf(acc0[r] + bs0) * w20;
                partial += tanhf(acc1[r] + bs1) * w21;
            }
        }
    }
    #pragma unroll
    for (int off = 16; off > 0; off >>= 1) partial += __shfl_xor(partial, off, 32);
    if (lane == 0) atomicAdd(&wsum[st], partial);
}

// ---------------------------------------------------------------------------
// Elementwise / edge kernels
// ---------------------------------------------------------------------------
__global__ void fill_k(float* p, int n, float v)
{
    int i = blockIdx.x * blockDim.x + threadIdx.x;
    if (i < n) p[i] = v;
}

DEV_INLINE void atomicMaxF(float* a, float v)
{
    // sign-split trick: correct IEEE float max for mixed signs
    if (v >= 0.f) atomicMax((int*)a, __float_as_int(v));
    else          atomicMin((unsigned int*)a, __float_as_uint(v));
}

// one wave per edge: per-head dot(q[dst], k[src]) over D, scatter-max into m
__global__ void edge_logits_k(const int* __restrict__ ei, const int* __restrict__ et,
                              const float* __restrict__ q, const float* __restrict__ kk,
                              float* __restrict__ logits, float* __restrict__ m, int t)
{
    int wid  = (blockIdx.x * blockDim.x + threadIdx.x) >> 5;
    int lane = threadIdx.x & 31;
    if (wid >= Ee) return;
    if (et[wid] != t) return;
    int src = ei[wid], dst = ei[Ee + wid];
    const float* qd = q  + (size_t)dst * (Hh * Dd);
    const float* ks = kk + (size_t)src * (Hh * Dd);
    #pragma unroll
    for (int h = 0; h < Hh; ++h) {
        float p = 0.f;
        #pragma unroll
        for (int i = 0; i < Dd / 32; ++i) {
            int c = h * Dd + lane + i * 32;
            p += qd[c] * ks[c];
        }
        #pragma unroll
        for (int off = 16; off > 0; off >>= 1) p += __shfl_xor(p, off, 32);
        p *= 0.08838834764831845f;              // 1/sqrt(D)
        if (lane == 0) { logits[wid * 2 + h] = p; atomicMaxF(&m[dst * 2 + h], p); }
    }
}

__global__ void edge_exp_k(const int* __restrict__ ei, const int* __restrict__ et,
                           const float* __restrict__ logits, const float* __restrict__ m,
                           float* __restrict__ eb, float* __restrict__ denom, int t)
{
    int idx = blockIdx.x * blockDim.x + threadIdx.x;
    if (idx >= Ee * Hh) return;
    int e = idx >> 1, h = idx & 1;
    if (et[e] != t) return;
    int dst = ei[Ee + e];
    float ex = __expf(logits[idx] - m[dst * 2 + h]);
    eb[idx] = ex;
    atomicAdd(&denom[dst * 2 + h], ex);
}

// one 256-thread block per edge: contiguous gather of v[src][0:256] and
// contiguous atomic-add into num[dst][0:256] (L2-resident)
__global__ void edge_scatter_k(const int* __restrict__ ei, const int* __restrict__ et,
                               const float* __restrict__ eb, const float* __restrict__ v,
                               float* __restrict__ num, int t)
{
    int e = blockIdx.x;
    if (et[e] != t) return;
    int j = threadIdx.x;                 // h*128 + d
    int src = ei[e], dst = ei[Ee + e];
    float ex = eb[e * 2 + (j >> 7)];
    atomicAdd(&num[(size_t)dst * 256 + j], ex * v[(size_t)src * 256 + j]);
}

// u = mean_h(num/denom) + skip   (skip result pre-stored in u)
__global__ void node_u_k(const float* __restrict__ num, const float* __restrict__ denom,
                         float* __restrict__ u)
{
    int idx = blockIdx.x * blockDim.x + threadIdx.x;
    if (idx >= Nn * Dd) return;
    int n = idx >> 7, dd = idx & 127;
    float d0 = fmaxf(denom[n * 2 + 0], 1e-16f);
    float d1 = fmaxf(denom[n * 2 + 1], 1e-16f);
    float mean = 0.5f * (num[(size_t)n * 256 + dd] / d0 +
                         num[(size_t)n * 256 + 128 + dd] / d1);
    u[idx] = mean + u[idx];
}

// z[:,t,:] = tanh(u)*sigmoid(gate) + x*(1-sigmoid(gate))
__global__ void z_write_k(const float* __restrict__ gate, const float* __restrict__ u,
                          const float* __restrict__ x, float* __restrict__ z, int t)
{
    int idx = blockIdx.x * blockDim.x + threadIdx.x;
    if (idx >= Nn * Dd) return;
    int n = idx >> 7, dd = idx & 127;
    float a = 1.f / (1.f + __expf(-gate[idx]));
    z[(size_t)n * 256 + t * 128 + dd] = tanhf(u[idx]) * a + x[idx] * (1.f - a);
}

// alpha[t] = (1/S) * sum_s softmax_t(wsum[s,:]/N)[t]
__global__ void alpha_k(const float* __restrict__ wsum, float* __restrict__ alpha)
{
    if (threadIdx.x == 0 && blockIdx.x == 0) {
        const float invN = 1.f / (float)Nn;
        float a0 = 0.f, a1 = 0.f;
        for (int s = 0; s < Ss; ++s) {
            float w0 = wsum[s * 2 + 0] * invN, w1 = wsum[s * 2 + 1] * invN;
            float mx = fmaxf(w0, w1);
            float e0 = __expf(w0 - mx), e1 = __expf(w1 - mx);
            float inv = 1.f / (e0 + e1);
            a0 += e0 * inv; a1 += e1 * inv;
        }
        alpha[0] = a0 / (float)Ss;
        alpha[1] = a1 / (float)Ss;
    }
}

// x = lrelu(alpha0 * z[:,0,:] + alpha1 * z[:,1,:])   (layer output, in place)
__global__ void combine_k(const float* __restrict__ z, const float* __restrict__ alpha,
                          float* __restrict__ x)
{
    int idx = blockIdx.x * blockDim.x + threadIdx.x;
    if (idx >= Nn * Dd) return;
    int n = idx >> 7, dd = idx & 127;
    float v = alpha[0] * z[(size_t)n * 256 + dd] +
              alpha[1] * z[(size_t)n * 256 + 128 + dd];
    x[idx] = v > 0.f ? v : 0.01f * v;
}

// final tiny projection to OUT2=2: one wave per node
__global__ void out2_k(const float* __restrict__ y, const float* __restrict__ Wout2,
                       const float* __restrict__ bout2, float* __restrict__ out)
{
    int wid  = (blockIdx.x * blockDim.x + threadIdx.x) >> 5;
    int lane = threadIdx.x & 31;
    if (wid >= Nn) return;
    const float* yr = y + (size_t)wid * Dd;
    float p0 = 0.f, p1 = 0.f;
    #pragma unroll
    for (int i = 0; i < Dd / 32; ++i) {
        int c = lane + i * 32;
        float yv = yr[c];
        p0 += yv * Wout2[c * 2 + 0];
        p1 += yv * Wout2[c * 2 + 1];
    }
    #pragma unroll
    for (int off = 16; off > 0; off >>= 1) {
        p0 += __shfl_xor(p0, off, 32);
        p1 += __shfl_xor(p1, off, 32);
    }
    if (lane == 0) { out[wid * 2 + 0] = p0 + bout2[0]; out[wid * 2 + 1] = p1 + bout2[1]; }
}

// ---------------------------------------------------------------------------
extern "C" void kernel_launch(void* const* d_in, const int* in_sizes, int n_in,
                              void* d_out, int out_size, void* d_ws, size_t ws_size,
                              hipStream_t stream)
{
    const float* features = (const float*)d_in[0];
    const int*   ei       = (const int*)  d_in[1];
    const int*   et       = (const int*)  d_in[2];
    const float* W1    = (const float*)d_in[3];
    const float* b1    = (const float*)d_in[4];
    const float* Wq    = (const float*)d_in[5];
    const float* bq    = (const float*)d_in[6];
    const float* Wk    = (const float*)d_in[7];
    const float* bk    = (const float*)d_in[8];
    const float* Wv    = (const float*)d_in[9];
    const float* bvv   = (const float*)d_in[10];
    const float* Wskip = (const float*)d_in[11];
    const float* bskip = (const float*)d_in[12];
    const float* Wg    = (const float*)d_in[13];
    const float* bg    = (const float*)d_in[14];
    const float* Wsem1 = (const float*)d_in[15];
    const float* bsem1 = (const float*)d_in[16];
    const float* wsem2 = (const float*)d_in[17];
    const float* Wout1 = (const float*)d_in[18];
    const float* bout1 = (const float*)d_in[19];
    const float* Wout2 = (const float*)d_in[20];
    const float* bout2 = (const float*)d_in[21];
    float* out = (float*)d_out;

    // workspace carve-out (~366 MB of f32)
    float* w = (float*)d_ws;
    size_t off = 0;
    auto alloc = [&](size_t n) { float* p = w + off; off += n; return p; };
    float* xbuf  = alloc((size_t)Nn * Dd);        // node state
    float* zbuf  = alloc((size_t)Nn * Tt * Dd);   // per-type gated outputs
    float* qbuf  = alloc((size_t)Nn * Hh * Dd);
    float* kbuf  = alloc((size_t)Nn * Hh * Dd);
    float* vbuf  = alloc((size_t)Nn * Hh * Dd);
    float* numb  = alloc((size_t)Nn * Hh * Dd);   // attention numerator
    float* ubuf  = alloc((size_t)Nn * Dd);        // skip -> u -> y1
    float* g2b   = alloc((size_t)Nn * Dd);        // x @ Wg[D:2D] + bg (per layer)
    float* gateb = alloc((size_t)Nn * Dd);
    float* logb  = alloc((size_t)Ee * Hh);
    float* eb    = alloc((size_t)Ee * Hh);
    float* mb    = alloc((size_t)Nn * Hh);
    float* denb  = alloc((size_t)Nn * Hh);
    float* wsum  = alloc(Ss * Tt);
    float* alph  = alloc(2);

    const dim3 gRows((Nn + 127) / 128, 0, 1);     // x-dim reused below

    auto gemm768_128 = [&](const float* A, const float* B, const float* bias,
                           const float* add, float* C, int act) {
        gemm_wmma_f32<768, 128><<<dim3((Nn + 127) / 128, 128 / 32), 256, 0, stream>>>(
            A, B, bias, add, C, Nn, act);
    };
    auto gemm128_256 = [&](const float* A, const float* B, const float* bias,
                           const float* add, float* C, int act) {
        gemm_wmma_f32<128, 256><<<dim3((Nn + 127) / 128, 256 / 32), 256, 0, stream>>>(
            A, B, bias, add, C, Nn, act);
    };
    auto gemm128_128 = [&](const float* A, const float* B, const float* bias,
                           const float* add, float* C, int act) {
        gemm_wmma_f32<128, 128><<<dim3((Nn + 127) / 128, 128 / 32), 256, 0, stream>>>(
            A, B, bias, add, C, Nn, act);
    };

    const int NE  = Nn * Dd;
    const int GNE = (NE + 255) / 256;

    // x = lrelu(features @ W1 + b1)
    gemm768_128(features, W1, b1, nullptr, xbuf, 1);

    for (int layer = 0; layer < 2; ++layer) {      // weight-shared layer, twice
        // x-half of the gate GEMM (shared across both edge types)
        gemm128_128(xbuf, Wg + Dd * Dd, bg, nullptr, g2b, 0);

        for (int t = 0; t < Tt; ++t) {
            gemm128_256(xbuf, Wq + (size_t)t * Dd * Hh * Dd, bq + t * Hh * Dd, nullptr, qbuf, 0);
            gemm128_256(xbuf, Wk + (size_t)t * Dd * Hh * Dd, bk + t * Hh * Dd, nullptr, kbuf, 0);
            gemm128_256(xbuf, Wv + (size_t)t * Dd * Hh * Dd, bvv + t * Hh * Dd, nullptr, vbuf, 0);

            fill_k<<<(Nn * Hh + 255) / 256, 256, 0, stream>>>(mb, Nn * Hh, -1e30f);
            hipMemsetAsync(denb, 0, (size_t)Nn * Hh * sizeof(float), stream);
            hipMemsetAsync(numb, 0, (size_t)Nn * Hh * Dd * sizeof(float), stream);

            edge_logits_k<<<(Ee + 7) / 8, 256, 0, stream>>>(ei, et, qbuf, kbuf, logb, mb, t);
            edge_exp_k<<<(Ee * Hh + 255) / 256, 256, 0, stream>>>(ei, et, logb, mb, eb, denb, t);
            edge_scatter_k<<<Ee, 256, 0, stream>>>(ei, et, eb, vbuf, numb, t);

            // skip: u <- x @ Wskip[t] + bskip[t], then u += mean_h(num/denom)
            gemm128_128(xbuf, Wskip + (size_t)t * Dd * Dd, bskip + t * Dd, nullptr, ubuf, 0);
            node_u_k<<<GNE, 256, 0, stream>>>(numb, denb, ubuf);

            // gate = u @ Wg[0:D] + g2  (bg already folded into g2)
            gemm128_128(ubuf, Wg, nullptr, g2b, gateb, 0);
            z_write_k<<<GNE, 256, 0, stream>>>(gateb, ubuf, xbuf, zbuf, t);
        }

        hipMemsetAsync(wsum, 0, Ss * Tt * sizeof(float), stream);
        sem_wsum_k<<<dim3((Nn + 127) / 128, Ss * Tt), 256, 0, stream>>>(zbuf, Wsem1, bsem1, wsem2, wsum);
        alpha_k<<<1, 32, 0, stream>>>(wsum, alph);
        combine_k<<<GNE, 256, 0, stream>>>(zbuf, alph, xbuf);
    }

    // y1 = lrelu(x @ Wout1 + bout1); out = y1 @ Wout2 + bout2
    gemm128_128(xbuf, Wout1, bout1, nullptr, ubuf, 1);
    out2_k<<<(Nn * 32 + 255) / 256, 256, 0, stream>>>(ubuf, Wout2, bout2, out);
}